// SparseTransformerConGT_8143257994114
// MI455X (gfx1250) — compile-verified
//
#include <hip/hip_runtime.h>
#include <math.h>

// Problem constants (match the reference)
#define NNODES  10000
#define DMODEL  128
#define NHEADS  8
#define CDIM    16
#define NLAYERS 2

typedef __attribute__((ext_vector_type(2))) float v2f;
typedef __attribute__((ext_vector_type(8))) float v8f;

// ---- ordered-float encoding so unsigned atomicMax == float max ----
__device__ __forceinline__ unsigned f32_key(float x) {
    unsigned b = __float_as_uint(x);
    return (b & 0x80000000u) ? ~b : (b | 0x80000000u);
}
__device__ __forceinline__ float key_f32(unsigned k) {
    unsigned b = (k & 0x80000000u) ? (k & 0x7FFFFFFFu) : ~k;
    return __uint_as_float(b);
}

// ============================================================================
// Fused 4-way projection GEMM: O[mid] = h @ W[mid] + B[mid]
// One wave32 per 16x16 output tile, K=128 consumed 4 at a time via
// v_wmma_f32_16x16x4_f32 (f32 in, f32 accumulate -> matches reference).
// grid.x = N/16 row tiles, grid.y = mid(4) * coltile(8)
// ============================================================================
__global__ void proj_wmma(const float* __restrict__ h,
                          const float* __restrict__ W0, const float* __restrict__ B0,
                          const float* __restrict__ W1, const float* __restrict__ B1,
                          const float* __restrict__ W2, const float* __restrict__ B2,
                          const float* __restrict__ W3, const float* __restrict__ B3,
                          float* __restrict__ O0, float* __restrict__ O1,
                          float* __restrict__ O2, float* __restrict__ O3)
{
    const int lane = threadIdx.x;            // 0..31 (wave32)
    const int m0   = blockIdx.x * 16;
    const int mid  = blockIdx.y >> 3;        // which weight matrix
    const int n0   = (blockIdx.y & 7) * 16;  // column tile

    const float* W; const float* B; float* O;
    switch (mid) {
        case 0:  W = W0; B = B0; O = O0; break;
        case 1:  W = W1; B = B1; O = O1; break;
        case 2:  W = W2; B = B2; O = O2; break;
        default: W = W3; B = B3; O = O3; break;
    }

    const int r  = lane & 15;   // A: M index, B/D: N index
    const int hh = lane >> 4;   // lane-half selects K (A/B) / M-half (D)

    // A fragment base: row m0+r, K offset 2*hh (pairs are contiguous -> b64 load)
    const float* arow = h + (size_t)(m0 + r) * DMODEL + 2 * hh;

    v8f acc = {};
    #pragma unroll
    for (int k = 0; k < DMODEL; k += 4) {
        v2f a = *(const v2f*)(arow + k);                       // A[m][k+2hh .. +1]
        v2f b;
        b.x = W[(size_t)(k + 2 * hh)     * DMODEL + n0 + r];   // B[k+2hh][n]
        b.y = W[(size_t)(k + 2 * hh + 1) * DMODEL + n0 + r];   // B[k+2hh+1][n]
        acc = __builtin_amdgcn_wmma_f32_16x16x4_f32(
                  /*neg_a=*/false, a, /*neg_b=*/false, b,
                  /*c_mod=*/(short)0, acc, /*reuse_a=*/false, /*reuse_b=*/false);
    }

    const float bias = B[n0 + r];
    #pragma unroll
    for (int i = 0; i < 8; ++i) {
        // D layout: VGPR i holds rows m0+i (lanes 0-15) / m0+i+8 (lanes 16-31)
        O[(size_t)(m0 + i + 8 * hh) * DMODEL + n0 + r] = acc[i] + bias;
    }
}

// ============================================================================
// Per-(edge,head) logits + segment max via ordered-uint atomicMax
// ============================================================================
__global__ void edge_logits(const float* __restrict__ q, const float* __restrict__ k,
                            const int* __restrict__ src, const int* __restrict__ dst,
                            int E, float* __restrict__ logits, unsigned* __restrict__ mkey)
{
    long long t = (long long)blockIdx.x * blockDim.x + threadIdx.x;
    if (t >= (long long)E * NHEADS) return;
    const int e  = (int)(t >> 3);
    const int hd = (int)(t & 7);
    const int s = src[e], d = dst[e];
    const float4* qp = (const float4*)(q + (size_t)d * DMODEL + hd * CDIM);
    const float4* kp = (const float4*)(k + (size_t)s * DMODEL + hd * CDIM);
    float acc = 0.f;
    #pragma unroll
    for (int i = 0; i < 4; ++i) {
        float4 a = qp[i], b = kp[i];
        acc += a.x * b.x + a.y * b.y + a.z * b.z + a.w * b.w;
    }
    acc *= 0.25f;  // 1/sqrt(C), C=16
    logits[t] = acc;
    atomicMax(&mkey[d * NHEADS + hd], f32_key(acc));
}

// ============================================================================
// exp(logit - max) in place + segment sum via f32 atomicAdd
// ============================================================================
__global__ void edge_exp(const int* __restrict__ dst, int E,
                         float* __restrict__ logits,
                         const unsigned* __restrict__ mkey, float* __restrict__ z)
{
    long long t = (long long)blockIdx.x * blockDim.x + threadIdx.x;
    if (t >= (long long)E * NHEADS) return;
    const int e  = (int)(t >> 3);
    const int hd = (int)(t & 7);
    const int d  = dst[e];
    const float m  = key_f32(mkey[d * NHEADS + hd]);
    const float ex = __expf(logits[t] - m);
    logits[t] = ex;
    atomicAdd(&z[d * NHEADS + hd], ex);
}

// ============================================================================
// agg[dst] += (e / z[dst]) * v[src]  -- one thread per (edge, feature)
// ============================================================================
__global__ void edge_agg(const float* __restrict__ v,
                         const int* __restrict__ src, const int* __restrict__ dst,
                         int E, const float* __restrict__ logits,
                         const float* __restrict__ z, float* __restrict__ agg)
{
    long long t = (long long)blockIdx.x * blockDim.x + threadIdx.x;
    if (t >= (long long)E * DMODEL) return;
    const int e  = (int)(t >> 7);
    const int j  = (int)(t & 127);
    const int hd = j >> 4;
    const int s = src[e], d = dst[e];
    const float alpha = logits[(size_t)e * NHEADS + hd] / z[d * NHEADS + hd];
    atomicAdd(&agg[(size_t)d * DMODEL + j], alpha * v[(size_t)s * DMODEL + j]);
}

// ============================================================================
// out = hin + agg + sout ; optional ReLU (layers 0..L-2)
// ============================================================================
__global__ void finalize(const float* __restrict__ hin, const float* __restrict__ agg,
                         const float* __restrict__ sout, float* __restrict__ hout,
                         int do_relu)
{
    int t = blockIdx.x * blockDim.x + threadIdx.x;
    if (t >= NNODES * DMODEL) return;
    float val = hin[t] + agg[t] + sout[t];
    if (do_relu) val = fmaxf(val, 0.f);
    hout[t] = val;
}

// ============================================================================
extern "C" void kernel_launch(void* const* d_in, const int* in_sizes, int n_in,
                              void* d_out, int out_size, void* d_ws, size_t ws_size,
                              hipStream_t stream)
{
    const float* x  = (const float*)d_in[0];
    const float* Wq = (const float*)d_in[1];
    const float* bq = (const float*)d_in[2];
    const float* Wk = (const float*)d_in[3];
    const float* bk = (const float*)d_in[4];
    const float* Wv = (const float*)d_in[5];
    const float* bv = (const float*)d_in[6];
    const float* Ws = (const float*)d_in[7];
    const float* bs = (const float*)d_in[8];
    const int*   aw = (const int*)d_in[9];

    const int E = in_sizes[9] / 2;
    const int* src = aw;       // attn_window[0]
    const int* dst = aw + E;   // attn_window[1]

    const size_t ND = (size_t)NNODES * DMODEL;
    const size_t NH = (size_t)NNODES * NHEADS;

    float* ws = (float*)d_ws;
    float* q      = ws;                 // ND
    float* kbuf   = q + ND;             // ND
    float* vbuf   = kbuf + ND;          // ND
    float* sout   = vbuf + ND;          // ND
    float* agg    = sout + ND;          // ND
    float* h1     = agg + ND;           // ND
    float* zbuf   = h1 + ND;            // NH
    unsigned* mkey = (unsigned*)(zbuf + NH);           // NH
    float* logits  = (float*)(mkey + NH);              // E*NHEADS

    float* outp = (float*)d_out;

    const dim3 gproj(NNODES / 16, 32);  // 625 row tiles x (4 mats * 8 col tiles)
    const long long EH = (long long)E * NHEADS;
    const long long ED = (long long)E * DMODEL;
    const int gEH = (int)((EH + 255) / 256);
    const int gED = (int)((ED + 255) / 256);
    const int gND = (int)((ND + 255) / 256);

    const float* hin = x;
    for (int l = 0; l < NLAYERS; ++l) {
        const size_t wofs = (size_t)l * DMODEL * DMODEL;
        const size_t bofs = (size_t)l * DMODEL;

        hipMemsetAsync(agg,  0, ND * sizeof(float), stream);
        hipMemsetAsync(zbuf, 0, NH * sizeof(float), stream);
        hipMemsetAsync(mkey, 0, NH * sizeof(unsigned), stream);

        proj_wmma<<<gproj, 32, 0, stream>>>(
            hin,
            Wq + wofs, bq + bofs,
            Wk + wofs, bk + bofs,
            Wv + wofs, bv + bofs,
            Ws + wofs, bs + bofs,
            q, kbuf, vbuf, sout);

        edge_logits<<<gEH, 256, 0, stream>>>(q, kbuf, src, dst, E, logits, mkey);
        edge_exp   <<<gEH, 256, 0, stream>>>(dst, E, logits, mkey, zbuf);
        edge_agg   <<<gED, 256, 0, stream>>>(vbuf, src, dst, E, logits, zbuf, agg);

        float* hout = (l == NLAYERS - 1) ? outp : h1;
        finalize<<<gND, 256, 0, stream>>>(hin, agg, sout, hout,
                                          (l < NLAYERS - 1) ? 1 : 0);
        hin = hout;
    }
}